// RoiLoss_32323923870248
// MI455X (gfx1250) — compile-verified
//
#include <hip/hip_runtime.h>
#include <hip/hip_bf16.h>
#include <math.h>

#define N_AGENTS 2048
#define R_CAND   128
#define NUM_MODS 6
#define T_STEPS  30
#define B_BATCH  16

typedef __attribute__((ext_vector_type(2))) float v2f;
typedef __attribute__((ext_vector_type(8))) float v8f;

static __device__ __forceinline__ float finf() { return __builtin_inff(); }

// ---------------------------------------------------------------------------
// Kernel 1: per-agent NMS + top-6 selection. One wave32 per agent.
// ---------------------------------------------------------------------------
__global__ __launch_bounds__(256) void k_select(
    const float* __restrict__ roi,    // (N*R, 5)
    const float* __restrict__ anch,   // (N*R, 4)
    const float* __restrict__ gt,     // (N, T, 2)
    float* __restrict__ ws_logit,     // (N, 6)
    float* __restrict__ ws_sel)       // (N, 6, 4)
{
    __shared__ float s_sc[8][R_CAND];
    __shared__ float s_gx[8][R_CAND];
    __shared__ float s_gy[8][R_CAND];
    __shared__ float s_g2[8][R_CAND];
    __shared__ float s_g3[8][R_CAND];
    __shared__ int   s_ord[8][R_CAND];
    __shared__ float s_kf[8][R_CAND];

    const int lane  = threadIdx.x & 31;
    const int wv    = threadIdx.x >> 5;
    const int agent = blockIdx.x * 8 + wv;

    float* sc = s_sc[wv];
    float* gx = s_gx[wv];
    float* gy = s_gy[wv];
    float* g2 = s_g2[wv];
    float* g3 = s_g3[wv];
    int*   od = s_ord[wv];
    float* kf = s_kf[wv];

    const long base = (long)agent * R_CAND;

    // Load candidates: score = roi[:,0]; goals = anchor + roi[:,1:5]
    #pragma unroll
    for (int k = 0; k < 4; ++k) {
        int c = lane + 32 * k;
        const float* rp = roi  + (base + c) * 5;
        const float* ap = anch + (base + c) * 4;
        sc[c] = rp[0];
        gx[c] = ap[0] + rp[1];
        gy[c] = ap[1] + rp[2];
        g2[c] = ap[2] + rp[3];
        g3[c] = ap[3] + rp[4];
    }
    __syncthreads();

    // Stable descending rank: rank = #{j: s_j > s_i} + #{j<i: s_j == s_i}
    #pragma unroll
    for (int k = 0; k < 4; ++k) {
        int i = lane + 32 * k;
        float si = sc[i];
        int rank = 0;
        for (int j = 0; j < R_CAND; ++j) {
            float sj = sc[j];
            rank += (sj > si) || ((sj == si) && (j < i));
        }
        od[rank] = i;
    }
    __syncthreads();

    // Each lane owns sorted slots lane, lane+32, lane+64, lane+96.
    float rgx[4], rgy[4];
    int oid[4];
    int kb[4];
    #pragma unroll
    for (int k = 0; k < 4; ++k) {
        int j = lane + 32 * k;
        oid[k] = od[j];
        rgx[k] = gx[oid[k]];
        rgy[k] = gy[oid[k]];
        kb[k]  = 0;
    }

    // Greedy NMS scan in sorted order. Boxes are 0.5x0.5 squares:
    // IoU > 0.5  <=>  inter > 1/6.
    #pragma unroll
    for (int g = 0; g < 4; ++g) {
        for (int ii = 0; ii < 32; ++ii) {
            const int i = g * 32 + ii;
            const int oi = od[i];
            const float bx = gx[oi];
            const float by = gy[oi];
            bool cond = false;
            #pragma unroll
            for (int k = 0; k < 4; ++k) {
                int j = lane + 32 * k;
                if ((j < i) && kb[k]) {
                    float wx = 0.5f - fabsf(rgx[k] - bx);
                    float wy = 0.5f - fabsf(rgy[k] - by);
                    float inter = fmaxf(wx, 0.f) * fmaxf(wy, 0.f);
                    cond = cond || (inter > (1.0f / 6.0f));
                }
            }
            const bool sup = (__any((int)cond) != 0);
            if (lane == ii) kb[g] = sup ? 0 : 1;
        }
    }

    // Keep count; fallback to keep-all if fewer than NUM_MODS survive.
    int cnt = 0;
    #pragma unroll
    for (int k = 0; k < 4; ++k) {
        unsigned long long b = __ballot(kb[k] != 0);
        cnt += __popcll(b);
    }
    const bool keepAll = (cnt < NUM_MODS);

    // Scatter keep flags back to original candidate order.
    #pragma unroll
    for (int k = 0; k < 4; ++k)
        kf[oid[k]] = (keepAll || kb[k]) ? 1.f : 0.f;
    __syncthreads();

    // L1 distance of goals to gt_preds[:, -1]; inf if not kept.
    const float gtx = gt[(long)agent * T_STEPS * 2 + (T_STEPS - 1) * 2 + 0];
    const float gty = gt[(long)agent * T_STEPS * 2 + (T_STEPS - 1) * 2 + 1];
    float dd[4];
    #pragma unroll
    for (int k = 0; k < 4; ++k) {
        int c = lane + 32 * k;
        float d = fabsf(gx[c] - gtx) + fabsf(gy[c] - gty);
        dd[k] = (kf[c] != 0.f) ? d : finf();
    }

    // Top-6 by ascending (dist, original index) — matches stable argsort.
    for (int m = 0; m < NUM_MODS; ++m) {
        float bv = finf();
        int   bi = R_CAND;
        #pragma unroll
        for (int k = 0; k < 4; ++k) {
            int c = lane + 32 * k;
            if (dd[k] < bv || (dd[k] == bv && c < bi)) { bv = dd[k]; bi = c; }
        }
        for (int off = 16; off > 0; off >>= 1) {
            float ov = __shfl_xor(bv, off, 32);
            int   oi = __shfl_xor(bi, off, 32);
            if (ov < bv || (ov == bv && oi < bi)) { bv = ov; bi = oi; }
        }
        if (lane == m) {
            ws_logit[(long)agent * NUM_MODS + m] = sc[bi];
            float* o = ws_sel + ((long)agent * NUM_MODS + m) * 4;
            o[0] = gx[bi]; o[1] = gy[bi]; o[2] = g2[bi]; o[3] = g3[bi];
        }
        #pragma unroll
        for (int k = 0; k < 4; ++k)
            if ((lane + 32 * k) == bi) dd[k] = finf();
    }
}

// ---------------------------------------------------------------------------
// Kernel 2: quadratic trajectory rollout via V_WMMA_F32_16X16X4_F32.
// D(16 rows x 16 timesteps) = A(16 coeff-triples x 4) * B(4 x 16 basis).
// rows r = agent*12 + mode*2 + coord; 24576 rows total, 16 per wave.
// ---------------------------------------------------------------------------
__global__ __launch_bounds__(256) void k_traj(
    const float* __restrict__ ctrs,   // (N, 2)
    const float* __restrict__ feats,  // (N, 20, 3)
    const float* __restrict__ ws_sel, // (N, 6, 4)
    float* __restrict__ ws_traj)      // (N*12, 30)
{
    const int lane   = threadIdx.x & 31;
    const int wv     = threadIdx.x >> 5;
    const int waveId = blockIdx.x * 8 + wv;
    const int rbase  = waveId * 16;
    const int l15    = lane & 15;
    const int r      = rbase + l15;

    const int agent = r / 12;
    const int rem   = r % 12;
    const int mode  = rem >> 1;
    const int coord = rem & 1;

    const float* sp = ws_sel + ((long)agent * NUM_MODS + mode) * 4;
    const float u = sp[coord];                       // sel0 / sel1
    const float v = sp[2 + coord];                   // sel2 / sel3
    const float p = ctrs[(long)agent * 2 + coord];   // ap0 / ap1
    const float q = feats[(long)agent * 60 + 19 * 3 + coord]; // ap2 / ap3

    const float c1 = (2.f * u * q + 2.f * p * q) / (2.f + q - v);
    const float c0 = u - p - c1;
    const float c2 = p;

    // A layout (16x4 f32): lanes 0-15 hold K=0,1; lanes 16-31 hold K=2,3.
    v2f A;
    A.x = (lane < 16) ? c0 : c2;
    A.y = (lane < 16) ? c1 : 0.f;

    #pragma unroll
    for (int h = 0; h < 2; ++h) {
        const int   tt = h * 16 + l15;
        const float s  = (float)tt * (1.f / (float)(T_STEPS - 1));
        // B layout (4x16 f32): lanes 0-15 hold K=0,1 rows; lanes 16-31 K=2,3.
        v2f Bv;
        Bv.x = (lane < 16) ? s * s : 1.f;
        Bv.y = (lane < 16) ? s     : 0.f;
        v8f C = {0.f, 0.f, 0.f, 0.f, 0.f, 0.f, 0.f, 0.f};
        v8f D = __builtin_amdgcn_wmma_f32_16x16x4_f32(
            /*neg_a=*/false, A, /*neg_b=*/false, Bv,
            /*c_mod=*/(short)0, C, /*reuse_a=*/false, /*reuse_b=*/false);

        if (tt < T_STEPS) {
            const int rowoff = (lane < 16) ? 0 : 8;
            #pragma unroll
            for (int m2 = 0; m2 < 8; ++m2) {
                const int rr = rbase + rowoff + m2;
                ws_traj[(long)rr * T_STEPS + tt] = D[m2];
            }
        }
    }
}

// ---------------------------------------------------------------------------
// Kernel 3: per-agent losses -> deterministic per-agent partials.
// ---------------------------------------------------------------------------
__global__ __launch_bounds__(256) void k_loss(
    const float* __restrict__ gt,            // (N, T, 2)
    const unsigned char* __restrict__ has,   // (N, T) bool
    const float* __restrict__ ws_logit,      // (N, 6)
    const float* __restrict__ ws_traj,       // (N*12, 30)
    float* __restrict__ part)                // [cls | reg | ncls | nreg] x N
{
    const int lane  = threadIdx.x & 31;
    const int agent = blockIdx.x * 8 + (threadIdx.x >> 5);
    const int t     = lane;

    // last = has + 0.1*t/T; argmax (first max), mask = max > 1
    const float hv = (t < T_STEPS) ? (has[(long)agent * T_STEPS + t] ? 1.f : 0.f) : 0.f;
    float mv = (t < T_STEPS) ? (hv + 0.1f * (float)t / (float)T_STEPS) : -finf();
    int   mi = t;
    for (int off = 16; off > 0; off >>= 1) {
        float ov = __shfl_xor(mv, off, 32);
        int   oi = __shfl_xor(mi, off, 32);
        if (ov > mv || (ov == mv && oi < mi)) { mv = ov; mi = oi; }
    }
    const int  li   = mi;
    const bool mask = (mv > 1.0f);

    const float gex = gt[(long)agent * 60 + li * 2 + 0];
    const float gey = gt[(long)agent * 60 + li * 2 + 1];

    // argmin over modes of endpoint distance (first min)
    float dv; int di;
    if (lane < NUM_MODS) {
        float px = ws_traj[((long)agent * 12 + lane * 2 + 0) * T_STEPS + li];
        float py = ws_traj[((long)agent * 12 + lane * 2 + 1) * T_STEPS + li];
        float dx = px - gex, dy = py - gey;
        dv = sqrtf(dx * dx + dy * dy);
        di = lane;
    } else { dv = finf(); di = lane; }
    for (int off = 16; off > 0; off >>= 1) {
        float ov = __shfl_xor(dv, off, 32);
        int   oi = __shfl_xor(di, off, 32);
        if (ov < dv || (ov == dv && oi < di)) { dv = ov; di = oi; }
    }
    const int minm = di;

    // BCE over 6 modes
    float bce = 0.f;
    if (lane < NUM_MODS) {
        float l = ws_logit[(long)agent * NUM_MODS + lane];
        float oneh = (lane == minm) ? 1.f : 0.f;
        bce = fmaxf(l, 0.f) - l * oneh + log1pf(expf(-fabsf(l)));
    }
    for (int off = 16; off > 0; off >>= 1) bce += __shfl_xor(bce, off, 32);

    // Smooth-L1 on best mode
    float reg = 0.f, wsum = 0.f;
    if (t < T_STEPS) {
        float bx = ws_traj[((long)agent * 12 + minm * 2 + 0) * T_STEPS + t];
        float by = ws_traj[((long)agent * 12 + minm * 2 + 1) * T_STEPS + t];
        float dx = bx - gt[(long)agent * 60 + t * 2 + 0];
        float dy = by - gt[(long)agent * 60 + t * 2 + 1];
        float ax = fabsf(dx), ay = fabsf(dy);
        float sx = (ax < 1.f) ? 0.5f * dx * dx : ax - 0.5f;
        float sy = (ay < 1.f) ? 0.5f * dy * dy : ay - 0.5f;
        float w  = (mask && hv > 0.f) ? 1.f : 0.f;
        reg  = w * (sx + sy);   // REG_COEF = 1
        wsum = w;
    }
    for (int off = 16; off > 0; off >>= 1) {
        reg  += __shfl_xor(reg,  off, 32);
        wsum += __shfl_xor(wsum, off, 32);
    }

    if (lane == 0) {
        part[0 * N_AGENTS + agent] = mask ? bce : 0.f;
        part[1 * N_AGENTS + agent] = reg;
        part[2 * N_AGENTS + agent] = mask ? 1.f : 0.f;
        part[3 * N_AGENTS + agent] = wsum;
    }
}

// ---------------------------------------------------------------------------
// Kernel 4: fixed-order reductions + traj_eval gather. Single block.
// ---------------------------------------------------------------------------
__global__ __launch_bounds__(256) void k_final(
    const float* __restrict__ part,
    const float* __restrict__ ws_traj,
    const int* __restrict__ natgs,    // (B,)
    float* __restrict__ out)          // 5764 floats
{
    __shared__ float red[256];
    const int tid = threadIdx.x;
    const int TRAJ_EVAL = B_BATCH * NUM_MODS * T_STEPS * 2; // 5760

    for (int q = 0; q < 4; ++q) {
        float s = 0.f;
        for (int i = tid; i < N_AGENTS; i += 256) s += part[q * N_AGENTS + i];
        red[tid] = s;
        __syncthreads();
        for (int off = 128; off > 0; off >>= 1) {
            if (tid < off) red[tid] += red[tid + off];
            __syncthreads();
        }
        if (tid == 0) {
            float v = red[0];
            if      (q == 0) out[0] = v;
            else if (q == 1) out[1] = v;
            else if (q == 2) out[2 + TRAJ_EVAL + 0] = v;
            else             out[2 + TRAJ_EVAL + 1] = v;
        }
        __syncthreads();
    }

    // traj_eval = pred_trajs[[0, natgs[0], natgs[1], ...]]
    for (int i = tid; i < TRAJ_EVAL; i += 256) {
        int b    = i / (NUM_MODS * T_STEPS * 2);
        int r2   = i % (NUM_MODS * T_STEPS * 2);
        int mode = r2 / (T_STEPS * 2);
        int r3   = r2 % (T_STEPS * 2);
        int tt   = r3 >> 1;
        int cc   = r3 & 1;
        int agent = (b == 0) ? 0 : natgs[b - 1];
        out[2 + i] = ws_traj[((long)agent * 12 + mode * 2 + cc) * T_STEPS + tt];
    }
}

// ---------------------------------------------------------------------------
extern "C" void kernel_launch(void* const* d_in, const int* in_sizes, int n_in,
                              void* d_out, int out_size, void* d_ws, size_t ws_size,
                              hipStream_t stream) {
    const float* roi   = (const float*)d_in[0];
    const float* anch  = (const float*)d_in[1];
    const float* ctrs  = (const float*)d_in[2];
    const float* feats = (const float*)d_in[3];
    const float* gt    = (const float*)d_in[4];
    const unsigned char* has = (const unsigned char*)d_in[5];
    const int*   natgs = (const int*)d_in[6];
    float* out = (float*)d_out;
    float* ws  = (float*)d_ws;

    float* ws_logit = ws;                                       // 12288 f
    float* ws_sel   = ws_logit + (long)N_AGENTS * NUM_MODS;     // 49152 f
    float* ws_traj  = ws_sel   + (long)N_AGENTS * NUM_MODS * 4; // 737280 f
    float* part     = ws_traj  + (long)N_AGENTS * 12 * T_STEPS; // 8192 f

    k_select<<<N_AGENTS / 8, 256, 0, stream>>>(roi, anch, gt, ws_logit, ws_sel);
    k_traj<<<(N_AGENTS * 12 / 16) / 8, 256, 0, stream>>>(ctrs, feats, ws_sel, ws_traj);
    k_loss<<<N_AGENTS / 8, 256, 0, stream>>>(gt, has, ws_logit, ws_traj, part);
    k_final<<<1, 256, 0, stream>>>(part, ws_traj, natgs, out);
}